// NTXent_SVM_71932112273960
// MI455X (gfx1250) — compile-verified
//
#include <hip/hip_runtime.h>
#include <math.h>

// ---------------- problem constants (from reference) ----------------
#define BSZ   4
#define NW    48
#define DD    128
#define NZ    (NW*NW)       // 2304
#define MAXIT 15
#define SIGMA 0.1f
#define CREG  0.1f
#define TAU   1.0f

// ---------------- WMMA vector types (gfx1250, wave32) ----------------
typedef __attribute__((ext_vector_type(2)))  float v2f;
typedef __attribute__((ext_vector_type(8)))  float v8f;

// ---------------- workspace layout (float offsets) ----------------
static const size_t OFF_S   = 0;                                  // [BSZ][NW][DD] S (f32, unit rows)
static const size_t OFF_Q   = OFF_S  + (size_t)BSZ*NW*DD;         // [BSZ][NW][DD] Q
static const size_t OFF_K   = OFF_Q  + (size_t)BSZ*NW*DD;         // [BSZ][NW][NW] K = S S^T
static const size_t OFF_CP  = OFF_K  + (size_t)BSZ*NW*NW;         // [BSZ][NW][NW] compat[s][q]
static const size_t OFF_Z   = OFF_CP + (size_t)BSZ*NW*NW;         // IPM state
static const size_t OFF_SS  = OFF_Z  + (size_t)BSZ*NZ;
static const size_t OFF_LAM = OFF_SS + (size_t)BSZ*NZ;
static const size_t OFF_NU  = OFF_LAM+ (size_t)BSZ*NZ;
static const size_t OFF_G   = OFF_NU + (size_t)BSZ*NW;            // rhs1
static const size_t OFF_R2  = OFF_G  + (size_t)BSZ*NZ;            // -A z
static const size_t OFF_MU  = OFF_R2 + (size_t)BSZ*NW;            // mu per batch
static const size_t OFF_U   = OFF_MU + (size_t)BSZ;               // [BSZ][a][NW] M_a^{-1} g_a
static const size_t OFF_X   = OFF_U  + (size_t)BSZ*NW*NW;         // [BSZ][a][NW][NW] M_a^{-1}
static const size_t OFF_PL  = OFF_X  + (size_t)BSZ*NW*NW*NW;      // partial loss
static const size_t OFF_PC  = OFF_PL + (size_t)BSZ;               // partial correct

// f32 WMMA helper: one 16x16 tile accum over K in steps of 4.
// A frag: lane m+16h, VGPR v <-> K = v + 2h.  B frag mirrored (lane n+16h).
__device__ __forceinline__ v8f wmma4_f32(v2f a, v2f b, v8f c) {
  return __builtin_amdgcn_wmma_f32_16x16x4_f32(false, a, false, b, (short)0, c, false, false);
}

// ============ 1. normalize rows of z + gather permuted support/query ============
__global__ void ng_norm_gather(const float* __restrict__ z,
                               const int* __restrict__ rsupp,
                               const int* __restrict__ rquery,
                               float* __restrict__ Sf, float* __restrict__ Qf) {
  int bid  = blockIdx.x;                 // 0 .. 2*BSZ*NW-1
  int role = bid / (BSZ*NW);
  int rem  = bid % (BSZ*NW);
  int b = rem / NW, i = rem % NW;
  int perm = (role == 0) ? rsupp[b*NW + i] : rquery[b*NW + i];
  int srcrow = (role == 0) ? (b*NW + perm) : (BSZ*NW + b*NW + perm);
  int d = threadIdx.x;                   // 128 == DD
  float v = z[(size_t)srcrow*DD + d];
  __shared__ float red[DD];
  red[d] = v*v;
  __syncthreads();
  for (int s = DD/2; s > 0; s >>= 1) { if (d < s) red[d] += red[d+s]; __syncthreads(); }
  float nv = v / sqrtf(red[0]);
  size_t dst = ((size_t)b*NW + i)*DD + d;
  if (role == 0) Sf[dst] = nv; else Qf[dst] = nv;
}

// ============ 2. K = S S^T and compat = S Q^T via v_wmma_f32_16x16x4_f32 ============
// grid = BSZ, blockDim = (32, 18): wave 0..8 -> K tiles, 9..17 -> compat tiles.
__global__ void ng_gemm_wmma(const float* __restrict__ Sf, const float* __restrict__ Qf,
                             float* __restrict__ Kmat, float* __restrict__ Cmat) {
  int b    = blockIdx.x;
  int wv   = threadIdx.y;      // 0..17
  int lane = threadIdx.x;      // 0..31
  bool isK = (wv < 9);
  int t  = isK ? wv : wv - 9;
  int ti = t / 3, tj = t % 3;
  const float* Abase = Sf + (size_t)b*NW*DD;
  const float* Bbase = (isK ? Sf : Qf) + (size_t)b*NW*DD;
  float* Obase = (isK ? Kmat : Cmat) + (size_t)b*NW*NW;
  int n15 = lane & 15;
  int h   = lane >> 4;
  const float* Arow = Abase + (size_t)(ti*16 + n15)*DD;   // row m (= lane&15, both halves)
  const float* Brow = Bbase + (size_t)(tj*16 + n15)*DD;   // col n
  v8f acc = {};
  #pragma unroll
  for (int k0 = 0; k0 < DD; k0 += 4) {
    v2f a  = *(const v2f*)(Arow + k0 + 2*h);  // K = k0 + 2h + {0,1}
    v2f bb = *(const v2f*)(Brow + k0 + 2*h);
    acc = wmma4_f32(a, bb, acc);
  }
  #pragma unroll
  for (int r = 0; r < 8; ++r)                 // C: VGPR r -> M = r + 8h, N = lane&15
    Obase[(size_t)(ti*16 + r + 8*h)*NW + tj*16 + n15] = acc[r];
}

// ============ 3. IPM state init: z=0, s=1, lam=1, nu=0 ============
__global__ void ng_init_state(float* zst, float* sst, float* lamst, float* nust) {
  int idx = blockIdx.x*blockDim.x + threadIdx.x;
  if (idx < BSZ*NZ) { zst[idx] = 0.f; sst[idx] = 1.f; lamst[idx] = 1.f; }
  if (idx < BSZ*NW) nust[idx] = 0.f;
}

// ============ 4. residuals; (K (x) I) z done with LDS -> WMMA -> LDS ============
// grid = BSZ, blockDim = (32, 9) = 288 threads (9 waves, one 16x16 tile each).
__global__ void ng_residual(const float* __restrict__ Kmat,
                            const float* __restrict__ zst, const float* __restrict__ sst,
                            const float* __restrict__ lamst, const float* __restrict__ nust,
                            const int* __restrict__ rsupp,
                            float* __restrict__ g, float* __restrict__ r2,
                            float* __restrict__ muv) {
  int b = blockIdx.x;
  int lane = threadIdx.x, wv = threadIdx.y;
  int tid = wv*32 + lane;                     // 0..287
  const float* Kb = Kmat + (size_t)b*NW*NW;
  const float* zb = zst + (size_t)b*NZ;
  const float* sb = sst + (size_t)b*NZ;
  const float* lb = lamst + (size_t)b*NZ;
  const float* nub = nust + (size_t)b*NW;
  __shared__ float Ksh[NW*NW];
  __shared__ float zsh[NZ];
  __shared__ float qzsh[NZ];
  __shared__ float red[288];
  __shared__ float muS;
  for (int idx = tid; idx < NW*NW; idx += 288) Ksh[idx] = Kb[idx];
  float p = 0.f;
  for (int idx = tid; idx < NZ; idx += 288) { zsh[idx] = zb[idx]; p += sb[idx]*lb[idx]; }
  red[tid] = p; __syncthreads();
  if (tid < 32) red[tid] += red[tid + 256];
  __syncthreads();
  for (int s = 128; s > 0; s >>= 1) { if (tid < s) red[tid] += red[tid+s]; __syncthreads(); }
  if (tid == 0) { muS = red[0] / (float)NZ; muv[b] = muS; }
  __syncthreads();
  // qz = K @ Z  (M=i, Kdim=j, N=a), one tile per wave, f32 WMMA from LDS
  {
    int ti = wv / 3, tj = wv % 3;
    int n15 = lane & 15, h = lane >> 4;
    const float* Arow = Ksh + (ti*16 + n15)*NW;
    v8f acc = {};
    #pragma unroll
    for (int k0 = 0; k0 < NW; k0 += 4) {
      v2f a, bb;
      a[0]  = Arow[k0 + 2*h];
      a[1]  = Arow[k0 + 2*h + 1];
      bb[0] = zsh[(k0 + 2*h    )*NW + tj*16 + n15];
      bb[1] = zsh[(k0 + 2*h + 1)*NW + tj*16 + n15];
      acc = wmma4_f32(a, bb, acc);
    }
    #pragma unroll
    for (int r = 0; r < 8; ++r)
      qzsh[(ti*16 + r + 8*h)*NW + tj*16 + n15] = acc[r];
  }
  __syncthreads();
  float mu = muS;
  for (int i = tid; i < NW; i += 288) {
    float sum = 0.f;
    for (int a = 0; a < NW; ++a) sum += zsh[i*NW + a];
    r2[b*NW + i] = -sum;
  }
  for (int idx = tid; idx < NZ; idx += 288) {
    int i = idx / NW, a = idx % NW;
    float qz = zsh[idx] + qzsh[idx];           // (K(x)I + I) z
    int lab = rsupp[b*NW + i];
    float pe = (a == lab) ? -1.f : 0.f;
    float he = (a == lab) ? CREG : 0.f;
    float sv = sb[idx], lv = lb[idx];
    float rz = qz + pe + lv + nub[i];
    float rs = zsh[idx] + sv - he;
    float rc = sv*lv - SIGMA*mu;
    g[(size_t)b*NZ + idx] = -rz + (rc - lv*rs)/sv;
  }
}

// ============ 5. per-slice: invert M_a = K + I + diag(lam_a/s_a); u_a = M_a^{-1} g_a ============
#define ACOLS 97
#define ASTR  98
__global__ void ng_slice_invert(const float* __restrict__ Kmat,
                                const float* __restrict__ sst, const float* __restrict__ lamst,
                                const float* __restrict__ g,
                                float* __restrict__ X, float* __restrict__ u) {
  int blk = blockIdx.x;
  int b = blk / NW, a = blk % NW;
  int tid = threadIdx.x;
  __shared__ float aug[NW*ASTR];   // [M | I | g_a]
  __shared__ float fcol[NW];
  __shared__ float pvS;
  const float* Kb = Kmat + (size_t)b*NW*NW;
  const float* sb = sst + (size_t)b*NZ;
  const float* lb = lamst + (size_t)b*NZ;
  const float* gb = g + (size_t)b*NZ;
  for (int idx = tid; idx < NW*ACOLS; idx += 256) {
    int i = idx / ACOLS, j = idx % ACOLS;
    float v;
    if (j < NW) {
      v = Kb[i*NW + j];
      if (i == j) v += 1.f + lb[i*NW + a]/sb[i*NW + a];
    } else if (j < 2*NW) {
      v = (j - NW == i) ? 1.f : 0.f;
    } else {
      v = gb[i*NW + a];
    }
    aug[i*ASTR + j] = v;
  }
  __syncthreads();
  for (int k = 0; k < NW; ++k) {        // Gauss-Jordan (M_a SPD, diag >= 2: no pivoting)
    if (tid == 0) pvS = 1.f / aug[k*ASTR + k];
    if (tid >= 32 && tid < 32 + NW) { int i = tid - 32; fcol[i] = aug[i*ASTR + k]; }
    __syncthreads();
    float pv = pvS;
    for (int j = tid; j < ACOLS; j += 256) aug[k*ASTR + j] *= pv;
    __syncthreads();
    for (int idx = tid; idx < NW*ACOLS; idx += 256) {
      int i = idx / ACOLS, j = idx % ACOLS;
      if (i != k) aug[i*ASTR + j] -= fcol[i]*aug[k*ASTR + j];
    }
    __syncthreads();
  }
  float* Xb = X + (((size_t)b*NW + a)*NW)*NW;
  float* ub = u + ((size_t)b*NW + a)*NW;
  for (int idx = tid; idx < NW*NW; idx += 256) {
    int i = idx / NW, j = idx % NW;
    Xb[i*NW + j] = aug[i*ASTR + NW + j];
  }
  for (int i = tid; i < NW; i += 256) ub[i] = aug[i*ASTR + 2*NW];
}

// ============ 6. nu-solve (W dnu = t - r2), dz, step length, state update ============
#define WCOLS 49
#define WSTR  50
__global__ void ng_solve_update(const float* __restrict__ X, const float* __restrict__ u,
                                const float* __restrict__ r2, const float* __restrict__ muv,
                                const int* __restrict__ rsupp,
                                float* __restrict__ zst, float* __restrict__ sst,
                                float* __restrict__ lamst, float* __restrict__ nust) {
  int b = blockIdx.x, tid = threadIdx.x;
  __shared__ float Waug[NW*WSTR];
  __shared__ float fcol[NW];
  __shared__ float pvS;
  __shared__ float dnu[NW];
  __shared__ float dzL[NZ];
  __shared__ float red[256];
  const float* Xb = X + (size_t)b*NW*NW*NW;
  const float* ub = u + (size_t)b*NW*NW;
  for (int idx = tid; idx < NW*NW; idx += 256) {    // W = sum_a M_a^{-1}
    int i = idx / NW, j = idx % NW;
    float s = 0.f;
    for (int a = 0; a < NW; ++a) s += Xb[((size_t)a*NW + i)*NW + j];
    Waug[i*WSTR + j] = s;
  }
  for (int i = tid; i < NW; i += 256) {             // rhs = sum_a u_a - r2
    float t = 0.f;
    for (int a = 0; a < NW; ++a) t += ub[a*NW + i];
    Waug[i*WSTR + NW] = t - r2[b*NW + i];
  }
  __syncthreads();
  for (int k = 0; k < NW; ++k) {
    if (tid == 0) pvS = 1.f / Waug[k*WSTR + k];
    if (tid >= 32 && tid < 32 + NW) { int i = tid - 32; fcol[i] = Waug[i*WSTR + k]; }
    __syncthreads();
    float pv = pvS;
    for (int j = tid; j < WCOLS; j += 256) Waug[k*WSTR + j] *= pv;
    __syncthreads();
    for (int idx = tid; idx < NW*WCOLS; idx += 256) {
      int i = idx / WCOLS, j = idx % WCOLS;
      if (i != k) Waug[i*WSTR + j] -= fcol[i]*Waug[k*WSTR + j];
    }
    __syncthreads();
  }
  for (int i = tid; i < NW; i += 256) dnu[i] = Waug[i*WSTR + NW];
  __syncthreads();
  float* zb = zst + (size_t)b*NZ;
  float* sb = sst + (size_t)b*NZ;
  float* lb = lamst + (size_t)b*NZ;
  float mu = muv[b];
  float amin = 1e12f;
  for (int idx = tid; idx < NZ; idx += 256) {       // dz_a = u_a - X_a dnu
    int i = idx / NW, a = idx % NW;
    const float* Xa = Xb + ((size_t)a*NW + i)*NW;
    float d = ub[a*NW + i];
    for (int j = 0; j < NW; ++j) d -= Xa[j]*dnu[j];
    dzL[idx] = d;
    int lab = rsupp[b*NW + i];
    float he = (a == lab) ? CREG : 0.f;
    float sv = sb[idx], lv = lb[idx], zv = zb[idx];
    float rs = zv + sv - he;
    float ds = -rs - d;
    float rc = sv*lv - SIGMA*mu;
    float dl = (-rc - lv*ds)/sv;
    if (ds < 0.f) amin = fminf(amin, -sv/ds);
    if (dl < 0.f) amin = fminf(amin, -lv/dl);
  }
  red[tid] = amin; __syncthreads();
  for (int s = 128; s > 0; s >>= 1) { if (tid < s) red[tid] = fminf(red[tid], red[tid+s]); __syncthreads(); }
  float alpha = 0.99f * fminf(1.f, red[0]);
  for (int idx = tid; idx < NZ; idx += 256) {
    int i = idx / NW, a = idx % NW;
    int lab = rsupp[b*NW + i];
    float he = (a == lab) ? CREG : 0.f;
    float sv = sb[idx], lv = lb[idx], zv = zb[idx];
    float d = dzL[idx];
    float rs = zv + sv - he;
    float ds = -rs - d;
    float rc = sv*lv - SIGMA*mu;
    float dl = (-rc - lv*ds)/sv;
    zb[idx] = zv + alpha*d;
    sb[idx] = sv + alpha*ds;
    lb[idx] = lv + alpha*dl;
  }
  for (int i = tid; i < NW; i += 256) nust[b*NW + i] += alpha*dnu[i];
}

// ============ 7. logits via f32 WMMA, log-softmax over q, loss + accuracy ============
// grid = BSZ, blockDim = (32, 9).  logits[q][w] = TAU * sum_s compat[s][q]*qp[s][w]
__global__ void ng_logits_loss(const float* __restrict__ zst, const float* __restrict__ Cmat,
                               const int* __restrict__ rquery,
                               float* __restrict__ ploss, float* __restrict__ pcorr) {
  int b = blockIdx.x;
  int lane = threadIdx.x, wv = threadIdx.y;
  int tid = wv*32 + lane;                     // 0..287
  __shared__ float lg[NW*NW];                 // [q][w]
  __shared__ float redl[288];
  __shared__ float redc[288];
  const float* zb = zst + (size_t)b*NZ;       // qp[s][w] = zb[s*NW+w]
  const float* cb = Cmat + (size_t)b*NW*NW;   // compat[s][q] = cb[s*NW+q]
  {
    int ti = wv / 3, tj = wv % 3;
    int n15 = lane & 15, h = lane >> 4;
    int q = ti*16 + n15;                      // A row m (A = compat^T)
    int w = tj*16 + n15;                      // B col n
    v8f acc = {};
    #pragma unroll
    for (int k0 = 0; k0 < NW; k0 += 4) {
      v2f a, bb;
      a[0]  = cb[(k0 + 2*h    )*NW + q];
      a[1]  = cb[(k0 + 2*h + 1)*NW + q];
      bb[0] = zb[(k0 + 2*h    )*NW + w];
      bb[1] = zb[(k0 + 2*h + 1)*NW + w];
      acc = wmma4_f32(a, bb, acc);
    }
    #pragma unroll
    for (int r = 0; r < 8; ++r)
      lg[(ti*16 + r + 8*h)*NW + tj*16 + n15] = TAU*acc[r];
  }
  __syncthreads();
  float lsum = 0.f, csum = 0.f;
  if (tid < NW) {                   // per w: log-softmax over q, pick q_labels[b,w]
    int w = tid;
    float m = -1e30f;
    for (int q = 0; q < NW; ++q) m = fmaxf(m, lg[q*NW + w]);
    float se = 0.f;
    for (int q = 0; q < NW; ++q) se += expf(lg[q*NW + w] - m);
    float lse = m + logf(se);
    int qlab = rquery[b*NW + w];
    lsum = lg[qlab*NW + w] - lse;
  }
  if (tid >= 64 && tid < 64 + NW) { // per q: argmax over w vs label
    int q = tid - 64;
    float best = lg[q*NW + 0]; int bi = 0;
    for (int w = 1; w < NW; ++w) { float v = lg[q*NW + w]; if (v > best) { best = v; bi = w; } }
    csum = (bi == rquery[b*NW + q]) ? 1.f : 0.f;
  }
  redl[tid] = lsum; redc[tid] = csum; __syncthreads();
  // contributions only at tid < 112, so a 256-wide tree is sufficient
  for (int s = 128; s > 0; s >>= 1) {
    if (tid < s) { redl[tid] += redl[tid+s]; redc[tid] += redc[tid+s]; }
    __syncthreads();
  }
  if (tid == 0) { ploss[b] = redl[0]; pcorr[b] = redc[0]; }
}

// ============ 8. finalize (loss, acc) -> d_out ============
__global__ void ng_finalize(const float* __restrict__ ploss, const float* __restrict__ pcorr,
                            float* __restrict__ out) {
  if (threadIdx.x == 0 && blockIdx.x == 0) {
    float l = 0.f, c = 0.f;
    for (int b = 0; b < BSZ; ++b) { l += ploss[b]; c += pcorr[b]; }
    out[0] = -l / (float)(BSZ*NW);
    out[1] =  c / (float)(BSZ*NW);
  }
}

// ---------------------------------------------------------------------------
extern "C" void kernel_launch(void* const* d_in, const int* in_sizes, int n_in,
                              void* d_out, int out_size, void* d_ws, size_t ws_size,
                              hipStream_t stream) {
  (void)in_sizes; (void)n_in; (void)out_size; (void)ws_size;
  const float* z      = (const float*)d_in[0];
  const int*   rsupp  = (const int*)d_in[1];
  const int*   rquery = (const int*)d_in[2];
  float* wsf = (float*)d_ws;

  float* Sf   = wsf + OFF_S;
  float* Qf   = wsf + OFF_Q;
  float* Kmat = wsf + OFF_K;
  float* Cmat = wsf + OFF_CP;
  float* zst  = wsf + OFF_Z;
  float* sst  = wsf + OFF_SS;
  float* lamst= wsf + OFF_LAM;
  float* nust = wsf + OFF_NU;
  float* g    = wsf + OFF_G;
  float* r2   = wsf + OFF_R2;
  float* muv  = wsf + OFF_MU;
  float* u    = wsf + OFF_U;
  float* X    = wsf + OFF_X;
  float* pl   = wsf + OFF_PL;
  float* pc   = wsf + OFF_PC;

  ng_norm_gather<<<2*BSZ*NW, DD, 0, stream>>>(z, rsupp, rquery, Sf, Qf);
  ng_gemm_wmma<<<BSZ, dim3(32, 18), 0, stream>>>(Sf, Qf, Kmat, Cmat);
  ng_init_state<<<(BSZ*NZ + 255)/256, 256, 0, stream>>>(zst, sst, lamst, nust);
  for (int it = 0; it < MAXIT; ++it) {
    ng_residual<<<BSZ, dim3(32, 9), 0, stream>>>(Kmat, zst, sst, lamst, nust, rsupp, g, r2, muv);
    ng_slice_invert<<<BSZ*NW, 256, 0, stream>>>(Kmat, sst, lamst, g, X, u);
    ng_solve_update<<<BSZ, 256, 0, stream>>>(X, u, r2, muv, rsupp, zst, sst, lamst, nust);
  }
  ng_logits_loss<<<BSZ, dim3(32, 9), 0, stream>>>(zst, Cmat, rquery, pl, pc);
  ng_finalize<<<1, 32, 0, stream>>>(pl, pc, (float*)d_out);
}